// MCR2Loss_41953240547841
// MI455X (gfx1250) — compile-verified
//
#include <hip/hip_runtime.h>
#include <hip/hip_bf16.h>
#include <math.h>

// ---------------- config ----------------
#define NCLS   32
#define DIM    128
#define S_SEG  8      // segments per class (parallelism for the gram pass)
#define CH     64     // rows per LDS chunk in gram kernel
#define ZSTR   132    // LDS row stride (floats), padded to spread banks
#define NB     128    // blocks for binning/scatter
#define GRAM_THREADS 512

typedef __attribute__((ext_vector_type(2))) float v2f;
typedef __attribute__((ext_vector_type(8))) float v8f;
typedef int v4i_t __attribute__((vector_size(16)));   // matches builtin's expected pointee

#define AS1 __attribute__((address_space(1)))
#define AS3 __attribute__((address_space(3)))

// ---- CDNA5 async global->LDS copy (ASYNCcnt-tracked), guarded fallback ----
#if defined(__has_builtin)
#if __has_builtin(__builtin_amdgcn_global_load_async_to_lds_b128)
#define HAVE_ASYNC_LDS 1
#endif
#endif

#if defined(__has_builtin)
#if __has_builtin(__builtin_amdgcn_s_wait_asynccnt)
#define WAIT_ASYNC(n) __builtin_amdgcn_s_wait_asynccnt(n)
#endif
#endif
#ifndef WAIT_ASYNC
#define WAIT_ASYNC(n) asm volatile("s_wait_asynccnt %0" ::"n"(n) : "memory")
#endif

__device__ __forceinline__ void async_copy16(const float* g, float* l) {
#ifdef HAVE_ASYNC_LDS
    // generic shared pointer: low 32 bits are the LDS byte address (ISA aperture rule)
    __builtin_amdgcn_global_load_async_to_lds_b128(
        (AS1 v4i_t*)(unsigned long long)(const void*)g,
        (AS3 v4i_t*)(unsigned)(unsigned long long)(void*)l, 0, 0);
#else
    *(float4*)l = *(const float4*)g;
#endif
}

// ---------------- pass 1a: per-block class histogram (deterministic) ----------------
__global__ void k_count(const int* __restrict__ label, int M, int rowsPerBlk,
                        int* __restrict__ blkCnt) {
    int lane = threadIdx.x;                  // 32 threads (one wave)
    int base = blockIdx.x * rowsPerBlk;
    int cnt = 0;
    for (int r = 0; r < rowsPerBlk; r += 32) {
        int i = base + r + lane;
        int lab = (i < M) ? label[i] : -1;
        #pragma unroll
        for (int j = 0; j < NCLS; ++j) {
            unsigned m = (unsigned)__ballot(lab == j);
            if (lane == j) cnt += __popc(m);
        }
    }
    blkCnt[blockIdx.x * NCLS + lane] = cnt;
}

// ---------------- pass 1b: exclusive scans (1 wave) ----------------
__global__ void k_offsets(int* __restrict__ blkCnt, int* __restrict__ offsets,
                          int* __restrict__ counts) {
    int j = threadIdx.x;                     // 32 threads
    int run = 0;
    for (int b = 0; b < NB; ++b) {
        int t = blkCnt[b * NCLS + j];
        blkCnt[b * NCLS + j] = run;          // becomes per-block exclusive start
        run += t;
    }
    counts[j] = run;
    __syncthreads();
    if (j == 0) {
        int off = 0;
        for (int c = 0; c < NCLS; ++c) { offsets[c] = off; off += counts[c]; }
        offsets[NCLS] = off;
    }
}

// ---------------- pass 1c: stable scatter -> perm (deterministic) ----------------
__global__ void k_scatter(const int* __restrict__ label, int M, int rowsPerBlk,
                          const int* __restrict__ blkCnt, const int* __restrict__ offsets,
                          int* __restrict__ perm) {
    int lane = threadIdx.x;                  // one wave
    int base = blockIdx.x * rowsPerBlk;
    int start = offsets[lane] + blkCnt[blockIdx.x * NCLS + lane]; // class==lane start
    int running = 0;
    unsigned ltmask = (lane == 0) ? 0u : ((1u << lane) - 1u);
    for (int r = 0; r < rowsPerBlk; r += 32) {
        int i = base + r + lane;
        int lab = (i < M) ? label[i] : -1;
        #pragma unroll
        for (int j = 0; j < NCLS; ++j) {
            unsigned m = (unsigned)__ballot(lab == j);
            int bj = __shfl(start, j) + __shfl(running, j);
            if (lab == j) {
                int rank = __popc(m & ltmask);
                perm[bj + rank] = i;
            }
            if (lane == j) running += __popc(m);
        }
    }
}

// ---------------- pass 2: per-class Gram via V_WMMA_F32_16X16X4_F32 ----------------
// grid = NCLS*S_SEG blocks, 512 threads (16 waves). Each wave owns a 2x2 block of
// 16x16 output tiles (4 accumulators, shared A/B operands). Rows gathered via perm
// into a double-buffered LDS chunk using async global->LDS loads.
__device__ __forceinline__ void load_chunk(const float* __restrict__ Z,
                                           const int* __restrict__ perm,
                                           int segStart, int nRows, int chunkBase,
                                           float* dstRow, int ldr, int ldc) {
    int idx = chunkBase + ldr;
    if (idx < nRows) {
        int g = perm[segStart + idx];
        const float* src = Z + (size_t)g * DIM + ldc;
        #pragma unroll
        for (int q = 0; q < 4; ++q) async_copy16(src + 4 * q, dstRow + 4 * q);
    } else {
        float4 zz = make_float4(0.f, 0.f, 0.f, 0.f);
        #pragma unroll
        for (int q = 0; q < 4; ++q) *(float4*)(dstRow + 4 * q) = zz;
    }
}

__global__ __launch_bounds__(GRAM_THREADS) void k_gram(const float* __restrict__ Z,
                                                       const int* __restrict__ perm,
                                                       const int* __restrict__ offsets,
                                                       const int* __restrict__ counts,
                                                       float* __restrict__ partials) {
    __shared__ float zt[2][CH][ZSTR];

    int cls = blockIdx.x / S_SEG;
    int seg = blockIdx.x % S_SEG;
    int cnt = counts[cls];
    int segLen = (cnt + S_SEG - 1) / S_SEG;
    int segStart = offsets[cls] + seg * segLen;
    int nRows = cnt - seg * segLen;
    if (nRows > segLen) nRows = segLen;
    if (nRows < 0) nRows = 0;

    int tid  = threadIdx.x;
    int w    = tid >> 5;
    int lane = tid & 31;
    int ln   = lane & 15;      // M/N index within tile
    int hi   = lane >> 4;      // selects K half (A/B) or M+8 half (C/D)
    int ta0  = (w >> 2) * 2;   // 2 row-tiles  {ta0, ta0+1}
    int tb0  = (w & 3) * 2;    // 2 col-tiles  {tb0, tb0+1}

    v8f acc00 = {}, acc01 = {}, acc10 = {}, acc11 = {};

    int ldr = tid >> 3;        // 0..63  row loaded by this thread
    int ldc = (tid & 7) * 16;  // 16 floats per thread (4 x b128)

    int nChunks = (nRows + CH - 1) / CH;

    if (nChunks > 0)
        load_chunk(Z, perm, segStart, nRows, 0, &zt[0][ldr][ldc], ldr, ldc);

    for (int c = 0; c < nChunks; ++c) {
        int cur = c & 1;
        bool pref = (c + 1) < nChunks;
        if (pref)
            load_chunk(Z, perm, segStart, nRows, (c + 1) * CH,
                       &zt[cur ^ 1][ldr][ldc], ldr, ldc);
        // allow up to 8 async loads in flight (in-order retirement); a chunk is 4 per
        // thread, so this buffer's loads are not guaranteed drained here.
        if (pref) { WAIT_ASYNC(8); } else { WAIT_ASYNC(0); }
        __syncthreads();

        const float (*zb)[ZSTR] = zt[cur];
        #pragma unroll
        for (int k0 = 0; k0 < CH; k0 += 4) {
            int r0 = k0 + 2 * hi;   // lanes 0-15: K=k0,k0+1 ; lanes 16-31: K=k0+2,k0+3
            v2f a0, a1, b0, b1;
            a0[0] = zb[r0][ta0 * 16 + ln];       a0[1] = zb[r0 + 1][ta0 * 16 + ln];
            a1[0] = zb[r0][ta0 * 16 + 16 + ln];  a1[1] = zb[r0 + 1][ta0 * 16 + 16 + ln];
            b0[0] = zb[r0][tb0 * 16 + ln];       b0[1] = zb[r0 + 1][tb0 * 16 + ln];
            b1[0] = zb[r0][tb0 * 16 + 16 + ln];  b1[1] = zb[r0 + 1][tb0 * 16 + 16 + ln];
            acc00 = __builtin_amdgcn_wmma_f32_16x16x4_f32(false, a0, false, b0,
                                                          (short)0, acc00, false, false);
            acc01 = __builtin_amdgcn_wmma_f32_16x16x4_f32(false, a0, false, b1,
                                                          (short)0, acc01, false, false);
            acc10 = __builtin_amdgcn_wmma_f32_16x16x4_f32(false, a1, false, b0,
                                                          (short)0, acc10, false, false);
            acc11 = __builtin_amdgcn_wmma_f32_16x16x4_f32(false, a1, false, b1,
                                                          (short)0, acc11, false, false);
        }
        __syncthreads();
    }

    // C/D layout: VGPR r, lane l -> (M = r + 8*(l>>4), N = l&15)
    float* out = partials + (size_t)blockIdx.x * (DIM * DIM);
    #pragma unroll
    for (int r = 0; r < 8; ++r) {
        int row0 = ta0 * 16 + r + 8 * hi;
        int row1 = row0 + 16;
        out[row0 * DIM + tb0 * 16 + ln]       = acc00[r];
        out[row0 * DIM + tb0 * 16 + 16 + ln]  = acc01[r];
        out[row1 * DIM + tb0 * 16 + ln]       = acc10[r];
        out[row1 * DIM + tb0 * 16 + 16 + ln]  = acc11[r];
    }
}

// ---------------- pass 3: deterministic reduction of segment partials ----------------
__global__ void k_reduce(const float* __restrict__ partials, float* __restrict__ cov) {
    int e = blockIdx.x * blockDim.x + threadIdx.x;   // 0..16383
    float total = 0.f;
    for (int j = 0; j < NCLS; ++j) {
        float s = 0.f;
        for (int seg = 0; seg < S_SEG; ++seg)
            s += partials[((size_t)(j * S_SEG + seg)) * (DIM * DIM) + e];
        cov[(size_t)j * (DIM * DIM) + e] = s;
        total += s;
    }
    cov[(size_t)NCLS * (DIM * DIM) + e] = total;     // global cov = sum of class covs
}

// ---------------- pass 4: Cholesky logdet of each 128x128 PD matrix ----------------
__global__ __launch_bounds__(128) void k_chol(const float* __restrict__ cov,
                                              float* __restrict__ logdets) {
    __shared__ float A[DIM][DIM];                    // 64 KB
    int i = threadIdx.x;
    const float* src = cov + (size_t)blockIdx.x * (DIM * DIM);
    for (int c = 0; c < DIM; ++c) A[i][c] = src[i * DIM + c];
    __syncthreads();
    float ld = 0.f;
    for (int k = 0; k < DIM; ++k) {
        if (i == k) A[k][k] = sqrtf(fmaxf(A[k][k], 1e-30f));
        __syncthreads();
        float dk = A[k][k];
        if (i == 0) ld += 2.f * logf(dk);
        if (i > k) A[i][k] /= dk;
        __syncthreads();
        if (i > k) {
            float lik = A[i][k];
            for (int j = k + 1; j <= i; ++j) A[i][j] -= lik * A[j][k];
        }
        __syncthreads();
    }
    if (i == 0) logdets[blockIdx.x] = ld;
}

// ---------------- pass 5: scalar combine ----------------
__global__ void k_final(const int* __restrict__ counts, const float* __restrict__ logdets,
                        int M, float* __restrict__ out) {
    if (threadIdx.x == 0 && blockIdx.x == 0) {
        const float eps = 0.01f;
        float m  = (float)M;
        float dd = (float)DIM;
        // logdet(c*X) = d*log(c) + logdet(X)
        float expd = 0.5f * (dd * logf(dd / (m * eps)) + logdets[NCLS]);
        float comp = 0.f;
        for (int j = 0; j < NCLS; ++j) {
            float mj = (float)counts[j];
            if (mj > 0.f) {
                float safe = fmaxf(mj, 1.f);
                comp += (mj / (2.f * m)) * (dd * logf(dd / (safe * eps)) + logdets[j]);
            }
        }
        out[0] = comp - expd;
    }
}

// ---------------- host-side launch ----------------
extern "C" void kernel_launch(void* const* d_in, const int* in_sizes, int n_in,
                              void* d_out, int out_size, void* d_ws, size_t ws_size,
                              hipStream_t stream) {
    (void)n_in; (void)out_size; (void)ws_size;

    const float* Z     = (const float*)d_in[0];
    const int*   label = (const int*)d_in[1];
    int M = in_sizes[1];

    char* ws = (char*)d_ws;
    int rowsPerBlk = ((M + NB * 32 - 1) / (NB * 32)) * 32;

    int*   blkCnt  = (int*)(ws + 0);               // NB*32 ints = 16 KB
    int*   offsets = (int*)(ws + 16384);           // 33 ints
    int*   counts  = (int*)(ws + 16640);           // 32 ints
    float* logdets = (float*)(ws + 16896);         // 33 floats
    int*   perm    = (int*)(ws + 17408);           // M ints
    size_t permEnd = 17408 + (size_t)M * 4;
    size_t pOff    = (permEnd + 255) & ~(size_t)255;
    float* partials = (float*)(ws + pOff);         // NCLS*S_SEG*128*128 floats (16 MB)
    size_t cOff    = pOff + (size_t)NCLS * S_SEG * DIM * DIM * 4;
    float* cov     = (float*)(ws + cOff);          // 33*128*128 floats (2.1 MB)

    k_count  <<<NB, 32, 0, stream>>>(label, M, rowsPerBlk, blkCnt);
    k_offsets<<<1, 32, 0, stream>>>(blkCnt, offsets, counts);
    k_scatter<<<NB, 32, 0, stream>>>(label, M, rowsPerBlk, blkCnt, offsets, perm);
    k_gram   <<<NCLS * S_SEG, GRAM_THREADS, 0, stream>>>(Z, perm, offsets, counts, partials);
    k_reduce <<<(DIM * DIM) / 256, 256, 0, stream>>>(partials, cov);
    k_chol   <<<NCLS + 1, 128, 0, stream>>>(cov, logdets);
    k_final  <<<1, 32, 0, stream>>>(counts, logdets, M, (float*)d_out);
}